// Attention_13855564497020
// MI455X (gfx1250) — compile-verified
//
#include <hip/hip_runtime.h>
#include <hip/hip_bf16.h>

typedef __attribute__((ext_vector_type(2))) float v2f;
typedef __attribute__((ext_vector_type(8))) float v8f;

#define BATCH 8
#define LSEQ  2048
#define DH    64
#define QT    16            // query rows per workgroup
#define SSTR  (LSEQ + 4)    // LDS row stride in floats (bank-conflict pad)

// Fused attention: one workgroup = one (batch, 16-query-row) tile.
// LDS strip holds scores -> exp -> normalized probabilities for the whole row.
__global__ __launch_bounds__(256) void attn_fused_f32wmma(
    const float* __restrict__ Q,
    const float* __restrict__ Cx,
    const unsigned char* __restrict__ mask,   // jnp bool -> 1 byte
    float* __restrict__ out,                  // [B, L, D]
    float* __restrict__ att)                  // [B, L, L]
{
    extern __shared__ float smem[];
    float* sP   = smem;                 // [QT][SSTR]  score/prob strip
    float* sRed = sP + QT * SSTR;       // [16][16]    softmax partials
    float* sMax = sRed + 256;           // [16]
    float* sSum = sMax + 16;            // [16]
    float* sAcc = sSum + 16;            // [8][8][32]  AV cross-wave partials

    const int tid  = threadIdx.x;
    const int wave = tid >> 5;
    const int lane = tid & 31;
    const int m    = lane & 15;         // row (A) / col (B) within 16-tile
    const int half = lane >> 4;         // upper/lower half-wave -> K pair select
    const int q0   = blockIdx.x * QT;
    const int bb   = blockIdx.y;

    // ---------------- Phase 1: S = Q @ C^T * (1/sqrt(D)), masked ----------------
    // A fragment (Q rows, 16x4 f32 layout): VGPR0 = K=2*half, VGPR1 = K=2*half+1.
    const float* qrow = Q + ((size_t)bb * LSEQ + q0 + m) * DH;
    v2f afrag[16];
#pragma unroll
    for (int i = 0; i < 16; ++i) {
        afrag[i].x = qrow[4 * i + 2 * half];
        afrag[i].y = qrow[4 * i + 2 * half + 1];
    }

    for (int kt = wave; kt < LSEQ / 16; kt += 8) {
        // B[k][n] = Cx[kt*16 + n][k]  (Q @ C^T)
        const float* crow = Cx + ((size_t)bb * LSEQ + kt * 16 + m) * DH;
        v8f acc = {};
#pragma unroll
        for (int i = 0; i < 16; ++i) {
            v2f bfrag;
            bfrag.x = crow[4 * i + 2 * half];
            bfrag.y = crow[4 * i + 2 * half + 1];
            acc = __builtin_amdgcn_wmma_f32_16x16x4_f32(
                false, afrag[i], false, bfrag, (short)0, acc, false, false);
        }
        // C/D layout: VGPR r -> row (r + 8*half), col = lane&15
        const int kcol = kt * 16 + m;
#pragma unroll
        for (int r = 0; r < 8; ++r) {
            const int mc = r + 8 * half;
            float v = acc[r] * 0.125f;  // 1/sqrt(64)
            if (mask[((size_t)bb * LSEQ + q0 + mc) * LSEQ + kcol])
                v = -__builtin_inff();
            sP[mc * SSTR + kcol] = v;
        }
    }
    __syncthreads();

    // ---------------- Phase 2: row softmax over the 16x2048 strip ----------------
    {
        const int r = tid >> 4, sub = tid & 15;
        float mx = -__builtin_inff();
        for (int j = sub; j < LSEQ; j += 16) mx = fmaxf(mx, sP[r * SSTR + j]);
        sRed[r * 16 + sub] = mx;
        __syncthreads();
        if (sub == 0) {
            float v = sRed[r * 16];
#pragma unroll
            for (int i = 1; i < 16; ++i) v = fmaxf(v, sRed[r * 16 + i]);
            sMax[r] = v;
        }
        __syncthreads();
        const float rowm = sMax[r];
        float sum = 0.f;
        for (int j = sub; j < LSEQ; j += 16) {
            const float e = __expf(sP[r * SSTR + j] - rowm);
            sP[r * SSTR + j] = e;
            sum += e;
        }
        sRed[r * 16 + sub] = sum;
        __syncthreads();
        if (sub == 0) {
            float v = 0.f;
#pragma unroll
            for (int i = 0; i < 16; ++i) v += sRed[r * 16 + i];
            sSum[r] = 1.0f / v;
        }
        __syncthreads();
    }

    // Normalize in LDS and stream att to HBM (coalesced: contiguous cols per 256 thds).
    {
        float* attb = att + ((size_t)bb * LSEQ + q0) * LSEQ;
        for (int idx = tid; idx < QT * LSEQ; idx += 256) {
            const int rr = idx >> 11;          // / 2048
            const int cc = idx & (LSEQ - 1);
            const float p = sP[rr * SSTR + cc] * sSum[rr];
            sP[rr * SSTR + cc] = p;
            attb[(size_t)rr * LSEQ + cc] = p;
        }
    }
    __syncthreads();

    // ---------------- Phase 3: out = P @ Cx  (16x2048 @ 2048x64) ----------------
    {
        const int ntile = wave & 3;            // which 16-wide column tile of D
        const int khalf = wave >> 2;           // split K range across wave pairs
        const int ncol  = ntile * 16 + m;
        const float* cb = Cx + (size_t)bb * LSEQ * DH;
        const int kbase = khalf * (LSEQ / 2);

        v8f acc = {};
        for (int k0 = 0; k0 < LSEQ / 2; k0 += 4) {
            const int k = kbase + k0 + 2 * half;
            v2f a, bfr;
            a.x   = sP[m * SSTR + k];
            a.y   = sP[m * SSTR + k + 1];
            bfr.x = cb[(size_t)k * DH + ncol];
            bfr.y = cb[(size_t)(k + 1) * DH + ncol];
            acc = __builtin_amdgcn_wmma_f32_16x16x4_f32(
                false, a, false, bfr, (short)0, acc, false, false);
        }
        // combine the two K-halves through LDS
        float* myAcc = sAcc + wave * 256;
#pragma unroll
        for (int r = 0; r < 8; ++r) myAcc[r * 32 + lane] = acc[r];
        __syncthreads();
        if (khalf == 0) {
            const float* other = sAcc + (wave + 4) * 256;
            float* ob = out + ((size_t)bb * LSEQ + q0) * DH;
#pragma unroll
            for (int r = 0; r < 8; ++r) {
                const int mc = r + 8 * half;
                ob[(size_t)mc * DH + ncol] = acc[r] + other[r * 32 + lane];
            }
        }
    }
}

extern "C" void kernel_launch(void* const* d_in, const int* in_sizes, int n_in,
                              void* d_out, int out_size, void* d_ws, size_t ws_size,
                              hipStream_t stream) {
    const float*         Q    = (const float*)d_in[0];
    const float*         Cx   = (const float*)d_in[1];
    const unsigned char* mask = (const unsigned char*)d_in[2];

    // d_out = [out (B*L*D) | att (B*L*L)] concatenated flat, both fp32.
    float* out = (float*)d_out;
    float* att = out + (size_t)BATCH * LSEQ * DH;

    const size_t smem_bytes =
        (size_t)(QT * SSTR + 256 + 16 + 16 + 8 * 256) * sizeof(float);

    dim3 grid(LSEQ / QT, BATCH);   // 128 x 8 workgroups
    dim3 block(256);               // 8 wave32
    attn_fused_f32wmma<<<grid, block, smem_bytes, stream>>>(Q, Cx, mask, out, att);
}